// Embedding_58969900974133
// MI455X (gfx1250) — compile-verified
//
#include <hip/hip_runtime.h>
#include <hip/hip_bf16.h>

// Embedding gather: out[row, :] = table[idx[row], :], row in [0, 32*4096), DIM=128.
// Bandwidth-bound: 512B/row via B128 loads/stores. Table (16MB) cached in L2 (RT),
// output (64MB) streamed with non-temporal stores so it never evicts the table.
//
// Per wave (wave32): ONE coalesced b32 load fetches the wave's 8 row indices
// (lanes 0..7), v_readlane broadcasts each to an SGPR, then all 8 row gathers
// issue as a clause before any store -> 2 memory latencies per wave instead of 8.

#define DIM 128
#define ROWS_PER_WAVE 8

typedef float v4f __attribute__((ext_vector_type(4)));

__global__ __launch_bounds__(256) void embed_gather_kernel(
    const int* __restrict__ idx,
    const float* __restrict__ table,
    float* __restrict__ out,
    int nrows)
{
    const int lane = threadIdx.x & 31;
    const int wave = (blockIdx.x * blockDim.x + threadIdx.x) >> 5;
    const int row0 = wave * ROWS_PER_WAVE;
    if (row0 >= nrows) return;

    if (row0 + ROWS_PER_WAVE <= nrows) {
        // Fast path: full group of 8 rows.
        // One coalesced load: lanes 0..7 hold idx[row0..row0+7].
        const int myidx = idx[row0 + (lane & (ROWS_PER_WAVE - 1))];

        v4f v[ROWS_PER_WAVE];
#pragma unroll
        for (int r = 0; r < ROWS_PER_WAVE; ++r) {
            // Broadcast row r's index into an SGPR -> scalar-base gather load.
            const int sid = __builtin_amdgcn_readlane(myidx, r);
            const v4f* __restrict__ src = (const v4f*)(table + (size_t)sid * DIM);
            v[r] = src[lane];                       // global_load_b128 (RT)
        }

        v4f* __restrict__ dst0 = (v4f*)(out + (size_t)row0 * DIM);
#pragma unroll
        for (int r = 0; r < ROWS_PER_WAVE; ++r) {
            // Streaming output: non-temporal global_store_b128 (TH=NT),
            // constant immediate offset r*512B off one scalar base.
            __builtin_nontemporal_store(v[r], dst0 + r * (DIM / 4) + lane);
        }
    } else {
        // Tail path (not taken for 131072 rows): per-row guarded gather.
        for (int r = 0; r < ROWS_PER_WAVE; ++r) {
            const int row = row0 + r;
            if (row < nrows) {
                const int sid = __builtin_amdgcn_readfirstlane(idx[row]);
                const v4f* __restrict__ src = (const v4f*)(table + (size_t)sid * DIM);
                v4f v = src[lane];
                v4f* __restrict__ dst = (v4f*)(out + (size_t)row * DIM);
                __builtin_nontemporal_store(v, dst + lane);
            }
        }
    }
}

extern "C" void kernel_launch(void* const* d_in, const int* in_sizes, int n_in,
                              void* d_out, int out_size, void* d_ws, size_t ws_size,
                              hipStream_t stream) {
    (void)n_in; (void)out_size; (void)d_ws; (void)ws_size;

    const int*   idx   = (const int*)d_in[0];    // [32, 4096] int32 token ids
    const float* table = (const float*)d_in[1];  // [32000, 128] float32
    float*       out   = (float*)d_out;          // [32, 4096, 128] float32

    const int nrows = in_sizes[0];               // 32 * 4096 = 131072

    const int waves   = (nrows + ROWS_PER_WAVE - 1) / ROWS_PER_WAVE;
    const long long threads = (long long)waves * 32;   // wave32
    const int block   = 256;                            // 8 waves per block
    const int grid    = (int)((threads + block - 1) / block);

    embed_gather_kernel<<<grid, block, 0, stream>>>(idx, table, out, nrows);
}